// ScaledDotProductAttention_3925600108948
// MI455X (gfx1250) — compile-verified
//
#include <hip/hip_runtime.h>
#include <hip/hip_bf16.h>

#define B_   2
#define H_   8
#define N_   384
#define D_   64
#define DH_  512            // D*H, contiguous innermost dim of pos_edges_hidden
#define JC_  64             // j-chunk for single-HBM-pass edge streaming
#define NCH_ (N_ / JC_)     // 6 chunks
#define SCALE 0.125f        // 1/sqrt(64)
#define NEGV (-1000000000.0f)

typedef __attribute__((ext_vector_type(2))) float v2f;
typedef __attribute__((ext_vector_type(8))) float v8f;

// ---------------------------------------------------------------------------
// Kernel A: qk[b,h,i,j] = sum_d Q[b,h,i,d]*K[b,h,j,d]   (raw, unscaled)
// One wave per 16x16 output tile, v_wmma_f32_16x16x4_f32 chained over K=64.
// ---------------------------------------------------------------------------
__global__ __launch_bounds__(256) void qk_gemm_kernel(
    const float* __restrict__ Q, const float* __restrict__ K,
    float* __restrict__ qk_out)
{
  const int wid  = (blockIdx.x * blockDim.x + threadIdx.x) >> 5;
  const int lane = threadIdx.x & 31;
  const int bh = wid / (24 * 24);          // 24x24 tiles of 16x16 per (b,h)
  const int r  = wid % (24 * 24);
  const int it = r / 24, jt = r % 24;
  if (bh >= B_ * H_) return;

  const int m  = lane & 15;
  const int kb = (lane < 16) ? 0 : 2;
  const float* Qrow = Q + ((size_t)bh * N_ + it * 16 + m) * D_;  // A: M=lane%16
  const float* Krow = K + ((size_t)bh * N_ + jt * 16 + m) * D_;  // B: N=lane%16

  v8f c = {0.f, 0.f, 0.f, 0.f, 0.f, 0.f, 0.f, 0.f};
  #pragma unroll
  for (int k0 = 0; k0 < D_; k0 += 4) {
    v2f a, b;
    a.x = Qrow[k0 + kb];     a.y = Qrow[k0 + kb + 1];
    b.x = Krow[k0 + kb];     b.y = Krow[k0 + kb + 1];
    c = __builtin_amdgcn_wmma_f32_16x16x4_f32(false, a, false, b,
                                              (short)0, c, false, false);
  }

  float* ob = qk_out + (size_t)bh * N_ * N_;
  const int col    = jt * 16 + m;
  const int rowadd = (lane < 16) ? 0 : 8;
  #pragma unroll
  for (int rr = 0; rr < 8; ++rr) {
    int row = it * 16 + rr + rowadd;
    ob[(size_t)row * N_ + col] = c[rr];
  }
}

// ---------------------------------------------------------------------------
// Wave-level softmax over a 384-entry LDS row (one wave = one head).
// Normalizes the row in place, writes it to gout, optionally stores 1/l.
// ---------------------------------------------------------------------------
__device__ __forceinline__ void row_softmax(float* row, int lane,
                                            float* __restrict__ gout,
                                            float* invl_slot)
{
  float m = -3.4e38f;
  for (int j = lane; j < N_; j += 32) m = fmaxf(m, row[j]);
  #pragma unroll
  for (int off = 16; off; off >>= 1) m = fmaxf(m, __shfl_xor(m, off, 32));
  float l = 0.f;
  for (int j = lane; j < N_; j += 32) { float e = expf(row[j] - m); row[j] = e; l += e; }
  #pragma unroll
  for (int off = 16; off; off >>= 1) l += __shfl_xor(l, off, 32);
  const float inv = 1.f / l;
  if (invl_slot && lane == 0) *invl_slot = inv;
  for (int j = lane; j < N_; j += 32) { float a = row[j] * inv; row[j] = a; gout[j] = a; }
}

// ---------------------------------------------------------------------------
// Kernel B: per (b,i) block, all 8 heads, SINGLE HBM pass over the edge row.
//  chunk loop (64 j's = 128 KB, re-read in phase c while L2-hot):
//   (a) s1[h,j] = sum_d Q[h,d]*E[i,j,d,h]  -> masked scoresi into s_lds
//   (b) per-head chunk max, online rescale factor, p = exp(s - m_run)
//   (c) cti accumulators: acc = acc*corr + sum_j p[h,j]*E[j,d,h]
//  end: exact softmax of full scoresi row -> attni (ws) ; cti = acc / l
// qk_attn buffer: read raw qk, overwritten with plain attn (disjoint rows).
// ---------------------------------------------------------------------------
__global__ __launch_bounds__(256) void edge_attn_kernel(
    const float* __restrict__ Q, const int* __restrict__ mask,
    const float* __restrict__ E, float* qk_attn,
    float* __restrict__ attni_ws, float* __restrict__ cti_out)
{
  __shared__ float Qs[DH_];            //  2 KB: Q row, layout dd*8+hh
  __shared__ float s_lds[H_ * N_];     // 12 KB: qk -> scoresi -> attni
  __shared__ float a_lds[H_ * N_];     // 12 KB: plain scores -> attn
  __shared__ int   mlds[N_];
  __shared__ float p_lds[JC_ * H_];    //  2 KB: chunk exp-weights, [j][h]
  __shared__ float corr_lds[H_];       // online rescale per head
  __shared__ float mrun_lds[H_];       // running max per head
  __shared__ float invl_lds[H_];       // final 1/l per head

  const int tid = threadIdx.x;
  const int b = blockIdx.x / N_;
  const int i = blockIdx.x % N_;
  const int wv = tid >> 5, lane = tid & 31;

  for (int j = tid; j < N_; j += 256) mlds[j] = mask[b * N_ + j];
  for (int c = tid; c < DH_; c += 256) {
    int hh = c & 7, dd = c >> 3;
    Qs[c] = Q[((size_t)(b * H_ + hh) * N_ + i) * D_ + dd];
  }
  if (tid < H_) mrun_lds[tid] = -3.0e38f;
  __syncthreads();

  // load qk rows for this (b,i); build plain masked scores
  for (int c = tid; c < H_ * N_; c += 256) {
    int h = c / N_, j = c % N_;
    float v = qk_attn[((size_t)(b * H_ + h) * N_ + i) * N_ + j];
    s_lds[c] = v;
    a_lds[c] = mlds[j] ? NEGV : v * SCALE;
  }
  __syncthreads();

  // plain attn (wave wv owns head wv); overwrite qk staging with attn output
  row_softmax(a_lds + wv * N_, lane,
              qk_attn + ((size_t)(b * H_ + wv) * N_ + i) * N_, nullptr);
  __syncthreads();

  const float* Erow = E + (size_t)(b * N_ + i) * N_ * DH_;
  const int subj = tid >> 2;        // phase a: 0..63  -> key within chunk
  const int sub  = tid & 3;         // phase a: 128-component slice
  const int cc   = tid * 2;         // phase c: 2 contiguous components
  const int hc   = cc & 7;          // even head 0,2,4,6
  const int ddc  = cc >> 3;
  float accx = 0.f, accy = 0.f;     // online cti accumulators

  for (int ch = 0; ch < NCH_; ++ch) {
    const int j0 = ch * JC_;
    // ---- phase a: s1 + masked scoresi for this chunk -----------------
    {
      const int j = j0 + subj;
      const float4* E4 = (const float4*)(Erow + (size_t)j * DH_ + sub * 128);
      const float4* Q4 = (const float4*)(Qs + sub * 128);
      if (ch + 1 < NCH_) {            // prefetch next chunk (same slice)
        const float* nx = Erow + (size_t)(j + JC_) * DH_ + sub * 128;
        __builtin_prefetch(nx,      0, 0);
        __builtin_prefetch(nx + 32, 0, 0);
        __builtin_prefetch(nx + 64, 0, 0);
        __builtin_prefetch(nx + 96, 0, 0);
      }
      float acc[8] = {0,0,0,0,0,0,0,0};
      #pragma unroll 8
      for (int lc = 0; lc < 32; ++lc) {
        float4 e = E4[lc];
        float4 q = Q4[lc];
        int h0 = (lc & 1) * 4;               // hh cycles 0..7 per float4 pair
        acc[h0+0] += e.x * q.x;  acc[h0+1] += e.y * q.y;
        acc[h0+2] += e.z * q.z;  acc[h0+3] += e.w * q.w;
      }
      #pragma unroll
      for (int h = 0; h < 8; ++h) {          // reduce the 4 slices
        acc[h] += __shfl_xor(acc[h], 1, 32);
        acc[h] += __shfl_xor(acc[h], 2, 32);
      }
      if (sub == 0) {
        const bool mm = mlds[j] != 0;
        #pragma unroll
        for (int h = 0; h < 8; ++h) {
          float v = s_lds[h * N_ + j];
          s_lds[h * N_ + j] = mm ? NEGV : (v + acc[h]) * SCALE;
        }
      }
    }
    __syncthreads();
    // ---- phase b: per-head chunk max, rescale factor, p-weights ------
    {
      float* row = s_lds + wv * N_ + j0;
      float s0 = row[lane], s1 = row[lane + 32];
      float cm = fmaxf(s0, s1);
      #pragma unroll
      for (int off = 16; off; off >>= 1) cm = fmaxf(cm, __shfl_xor(cm, off, 32));
      float mold = mrun_lds[wv];
      float mnew = fmaxf(mold, cm);
      if (lane == 0) { mrun_lds[wv] = mnew; corr_lds[wv] = expf(mold - mnew); }
      p_lds[lane * H_ + wv]        = expf(s0 - mnew);   // [j][h] layout
      p_lds[(lane + 32) * H_ + wv] = expf(s1 - mnew);
    }
    __syncthreads();
    // ---- phase c: rescale + accumulate cti over chunk (L2-hot re-read)
    {
      accx *= corr_lds[hc];
      accy *= corr_lds[hc + 1];
      const float* Ep = Erow + (size_t)j0 * DH_ + cc;
      for (int jj = 0; jj < JC_; ++jj) {
        float2 e = *(const float2*)(Ep + (size_t)jj * DH_);
        accx += p_lds[jj * H_ + hc]     * e.x;
        accy += p_lds[jj * H_ + hc + 1] * e.y;
      }
    }
    __syncthreads();
  }

  // exact softmax of full edge-score row -> attni (LDS + ws), capture 1/l
  row_softmax(s_lds + wv * N_, lane,
              attni_ws + ((size_t)(b * H_ + wv) * N_ + i) * N_,
              &invl_lds[wv]);
  __syncthreads();

  // cti accumulators are referenced to the final running max == global max
  cti_out[((size_t)(b * H_ + hc)     * N_ + i) * D_ + ddc] = accx * invl_lds[hc];
  cti_out[((size_t)(b * H_ + hc + 1) * N_ + i) * D_ + ddc] = accy * invl_lds[hc + 1];
}

// ---------------------------------------------------------------------------
// Kernel C: context = attn @ V ; contexti += attni @ V   (WMMA fp32, K=384)
// One wave per 16x16 output tile; V tile loads shared between both GEMMs.
// ---------------------------------------------------------------------------
__global__ __launch_bounds__(256) void av_gemm_kernel(
    const float* __restrict__ V, const float* __restrict__ attn,
    const float* __restrict__ attni, float* __restrict__ ctx_out,
    float* __restrict__ cti_out)
{
  const int wid  = (blockIdx.x * blockDim.x + threadIdx.x) >> 5;
  const int lane = threadIdx.x & 31;
  const int bh = wid / 96;                  // 24 (i-tiles) x 4 (d-tiles)
  const int r  = wid % 96;
  const int it = r >> 2, jt = r & 3;
  if (bh >= B_ * H_) return;

  const int m  = lane & 15;
  const int kb = (lane < 16) ? 0 : 2;
  const float* Arow  = attn  + ((size_t)bh * N_ + it * 16 + m) * N_;
  const float* Irow  = attni + ((size_t)bh * N_ + it * 16 + m) * N_;
  const float* Vbase = V + (size_t)bh * N_ * D_ + jt * 16 + m;

  v8f cA = {0.f,0.f,0.f,0.f,0.f,0.f,0.f,0.f};
  v8f cI = {0.f,0.f,0.f,0.f,0.f,0.f,0.f,0.f};
  for (int k0 = 0; k0 < N_; k0 += 4) {
    const int k = k0 + kb;
    v2f a, ai, bv;
    a.x  = Arow[k];            a.y  = Arow[k + 1];
    ai.x = Irow[k];            ai.y = Irow[k + 1];
    bv.x = Vbase[(size_t)k * D_];
    bv.y = Vbase[(size_t)(k + 1) * D_];
    cA = __builtin_amdgcn_wmma_f32_16x16x4_f32(false, a,  false, bv,
                                               (short)0, cA, false, false);
    cI = __builtin_amdgcn_wmma_f32_16x16x4_f32(false, ai, false, bv,
                                               (short)0, cI, false, false);
  }

  const int col    = jt * 16 + m;
  const int rowadd = (lane < 16) ? 0 : 8;
  #pragma unroll
  for (int rr = 0; rr < 8; ++rr) {
    int row = it * 16 + rr + rowadd;
    size_t idx = ((size_t)bh * N_ + row) * D_ + col;
    ctx_out[idx] = cA[rr];
    cti_out[idx] += cI[rr];      // add edge part written by edge_attn_kernel
  }
}

// ---------------------------------------------------------------------------
extern "C" void kernel_launch(void* const* d_in, const int* in_sizes, int n_in,
                              void* d_out, int out_size, void* d_ws, size_t ws_size,
                              hipStream_t stream)
{
  const float* Q   = (const float*)d_in[0];
  const float* K   = (const float*)d_in[1];
  const float* V   = (const float*)d_in[2];
  const int*   msk = (const int*)  d_in[3];
  const float* E   = (const float*)d_in[4];
  // d_in[5] (spanmask_deges_hidden) unused by reference

  float* out   = (float*)d_out;
  float* cti   = out;                           // contexti: B*H*N*D = 393216
  float* ctx   = out + (size_t)B_*H_*N_*D_;     // context
  float* attn  = out + (size_t)2*B_*H_*N_*D_;   // attn: B*H*N*N (also qk staging)
  float* attni = (float*)d_ws;                  // B*H*N*N floats = 9.4 MB

  // A: raw qk staged into the attn output region (each edge_attn block reads
  //    exactly the rows it later overwrites with attn -> no hazard).
  qk_gemm_kernel<<<(B_*H_*24*24)/8, 256, 0, stream>>>(Q, K, attn);

  // B: per-(b,i) single-pass edge streaming + both softmaxes + edge contexti
  edge_attn_kernel<<<B_*N_, 256, 0, stream>>>(Q, msk, E, attn, attni, cti);

  // C: attn@V and attni@V (accumulates onto contexti)
  av_gemm_kernel<<<(B_*H_*24*4)/8, 256, 0, stream>>>(V, attn, attni, ctx, cti);
}